// DLRM_Net_12214886989971
// MI455X (gfx1250) — compile-verified
//
#include <hip/hip_runtime.h>
#include <math.h>

// DLRM fused forward for MI455X (gfx1250, wave32).
// Memory-bound (~8 MB total traffic, ~0.4us at 23.3 TB/s); single fused pass.
// Top layer-1 GEMM ([B,54]@[54,4]) runs on V_WMMA_F32_16X16X4_F32.

typedef __attribute__((ext_vector_type(2))) float v2f;
typedef __attribute__((ext_vector_type(8))) float v8f;

#define B_TOTAL   16384
#define N_DENSE   13
#define N_TABLES  26
#define VOCAB     1000000
#define EMB_DIM   2
#define K_TOP     54      // 2 dense-MLP outputs + 26*2 embedding values
#define K_PAD     56      // padded to multiple of 4 for 16x16x4 WMMA
#define WAVES_PB  8
#define ROWS_PW   16
#define X_STRIDE  57      // odd stride: lanes m=0..15 hit distinct LDS banks

__global__ __launch_bounds__(256) void dlrm_fused_kernel(
    const float* __restrict__ x_dense,
    const int*   __restrict__ x_cat,
    const float* __restrict__ emb,
    const float* __restrict__ bot_w1, const float* __restrict__ bot_b1,
    const float* __restrict__ bot_w2, const float* __restrict__ bot_b2,
    const float* __restrict__ top_w1, const float* __restrict__ top_b1,
    const float* __restrict__ top_w2, const float* __restrict__ top_b2,
    const float* __restrict__ top_w3, const float* __restrict__ top_b3,
    float* __restrict__ out)
{
    __shared__ float lds_w1[K_PAD * 16];                     // padded B tile (rows>=54, cols>=4 are 0)
    __shared__ float lds_x [WAVES_PB][ROWS_PW * X_STRIDE];   // per-wave A tile [16 x 56]
    __shared__ float lds_h [WAVES_PB][ROWS_PW * 4];          // per-wave layer-1 result [16 x 4]

    const int tid  = threadIdx.x;
    const int wave = tid >> 5;
    const int lane = tid & 31;
    const int m    = lane & 15;   // row within tile (also N index for B/C fragments)
    const int half = lane >> 4;   // 0: lanes 0-15, 1: lanes 16-31
    const int r0   = (blockIdx.x * WAVES_PB + wave) * ROWS_PW;

    // Stage zero-padded top_w1 as a 56x16 tile once per block.
    for (int i = tid; i < K_PAD * 16; i += 256) {
        int row = i >> 4, col = i & 15;
        lds_w1[i] = (row < K_TOP && col < 4) ? top_w1[row * 4 + col] : 0.0f;
    }

    float* xrow = &lds_x[wave][0];

    // ---- Phase 1: bottom MLP 13->3->2 (scalar, one row per lane 0-15) ----
    if (half == 0) {
        float xd[N_DENSE];
        #pragma unroll
        for (int i = 0; i < N_DENSE; ++i) xd[i] = x_dense[(r0 + m) * N_DENSE + i];
        float h1[3];
        #pragma unroll
        for (int j = 0; j < 3; ++j) {
            float s = bot_b1[j];
            #pragma unroll
            for (int i = 0; i < N_DENSE; ++i) s = fmaf(xd[i], bot_w1[i * 3 + j], s);
            h1[j] = fmaxf(s, 0.0f);
        }
        #pragma unroll
        for (int k = 0; k < 2; ++k) {
            float s = bot_b2[k];
            #pragma unroll
            for (int j = 0; j < 3; ++j) s = fmaf(h1[j], bot_w2[j * 2 + k], s);
            xrow[m * X_STRIDE + k] = fmaxf(s, 0.0f);
        }
        xrow[m * X_STRIDE + 54] = 0.0f;   // K padding
        xrow[m * X_STRIDE + 55] = 0.0f;
    }

    // ---- Phase 2: 26 embedding gathers (2 tables x 16 rows per step) ----
    #pragma unroll
    for (int t2 = 0; t2 < N_TABLES / 2; ++t2) {
        const int t   = t2 * 2 + half;
        int idx = x_cat[(r0 + m) * N_TABLES + t];
        idx = idx < 0 ? 0 : (idx >= VOCAB ? VOCAB - 1 : idx);
        const float* src = emb + ((size_t)t * VOCAB + (size_t)idx) * EMB_DIM;
        const float e0 = src[0];
        const float e1 = src[1];
        xrow[m * X_STRIDE + 2 + 2 * t]     = e0;
        xrow[m * X_STRIDE + 2 + 2 * t + 1] = e1;
    }

    __syncthreads();

    // ---- Phase 3: [16x54] @ [54x4] via 14x V_WMMA_F32_16X16X4_F32 ----
    // A 16x4 layout: lanes 0-15 hold (M=lane, K=k0..k0+1), lanes 16-31 hold K=k0+2..k0+3.
    // B 4x16 layout mirrors A; C/D 16x16: VGPR r = row r (lanes 0-15) / r+8 (lanes 16-31).
    v8f acc = {};
    const int koff = half * 2;
    #pragma unroll
    for (int k0 = 0; k0 < K_PAD; k0 += 4) {
        v2f a, b;
        a[0] = xrow[m * X_STRIDE + k0 + koff];
        a[1] = xrow[m * X_STRIDE + k0 + koff + 1];
        b[0] = lds_w1[(k0 + koff)     * 16 + m];
        b[1] = lds_w1[(k0 + koff + 1) * 16 + m];
        acc = __builtin_amdgcn_wmma_f32_16x16x4_f32(
                  /*neg_a=*/false, a, /*neg_b=*/false, b,
                  /*c_mod=*/(short)0, acc, /*reuse_a=*/false, /*reuse_b=*/false);
    }

    // Scatter the 4 useful output columns (N = 0..3) to LDS.
    if (m < 4) {
        #pragma unroll
        for (int r = 0; r < 8; ++r)
            lds_h[wave][(half * 8 + r) * 4 + m] = acc[r];
    }
    __syncthreads();

    // ---- Phase 4: bias+relu, 4->2->1, sigmoid (one row per lane 0-15) ----
    if (half == 0) {
        float h[4], g[2];
        #pragma unroll
        for (int j = 0; j < 4; ++j)
            h[j] = fmaxf(lds_h[wave][m * 4 + j] + top_b1[j], 0.0f);
        #pragma unroll
        for (int k = 0; k < 2; ++k) {
            float s = top_b2[k];
            #pragma unroll
            for (int j = 0; j < 4; ++j) s = fmaf(h[j], top_w2[j * 2 + k], s);
            g[k] = fmaxf(s, 0.0f);
        }
        const float z = fmaf(g[0], top_w3[0], fmaf(g[1], top_w3[1], top_b3[0]));
        out[r0 + m] = 1.0f / (1.0f + __expf(-z));
    }
}

extern "C" void kernel_launch(void* const* d_in, const int* in_sizes, int n_in,
                              void* d_out, int out_size, void* d_ws, size_t ws_size,
                              hipStream_t stream) {
    (void)in_sizes; (void)n_in; (void)d_ws; (void)ws_size; (void)out_size;
    const float* x_dense = (const float*)d_in[0];
    const int*   x_cat   = (const int*)  d_in[1];
    const float* emb     = (const float*)d_in[2];
    const float* bot_w1  = (const float*)d_in[3];
    const float* bot_b1  = (const float*)d_in[4];
    const float* bot_w2  = (const float*)d_in[5];
    const float* bot_b2  = (const float*)d_in[6];
    const float* top_w1  = (const float*)d_in[7];
    const float* top_b1  = (const float*)d_in[8];
    const float* top_w2  = (const float*)d_in[9];
    const float* top_b2  = (const float*)d_in[10];
    const float* top_w3  = (const float*)d_in[11];
    const float* top_b3  = (const float*)d_in[12];
    float* out = (float*)d_out;

    const int rows_per_block = WAVES_PB * ROWS_PW;           // 128
    const int grid = B_TOTAL / rows_per_block;               // 128 blocks, 1024 waves
    dlrm_fused_kernel<<<grid, 256, 0, stream>>>(
        x_dense, x_cat, emb,
        bot_w1, bot_b1, bot_w2, bot_b2,
        top_w1, top_b1, top_w2, top_b2, top_w3, top_b3,
        out);
}